// VectorQuantize2_35923106464026
// MI455X (gfx1250) — compile-verified
//
#include <hip/hip_runtime.h>

typedef __attribute__((ext_vector_type(2))) float v2f;
typedef __attribute__((ext_vector_type(8))) float v8f;

#define M_TOT 8192
#define N_TOT 16384
#define KDIM  256
#define MT    64           // z rows per workgroup
#define NCH   128          // codebook rows per chunk (workgroup N tile)
#define NSPLIT 2           // split the code sweep across gridDim.y
#define CHUNKS_PER_SPLIT ((N_TOT / NCH) / NSPLIT)   // 64
#define ASTRIDE (KDIM + 4) // 260 floats: stride%64banks==4 -> conflict-free b64 frag loads

#define AS_BYTES (MT * ASTRIDE * 4)    // 66560
#define BS_BYTES (NCH * ASTRIDE * 4)   // 133120
#define RED_BYTES (8 * 2 * 8 * 32 * 4) // 16384

// ---------------------------------------------------------------------------
// Kernel 1: pack z [8,256,32,32] -> zf [8192,256]   (n = b*1024 + h*32 + w)
// ---------------------------------------------------------------------------
__global__ __launch_bounds__(256) void pack_z_kernel(const float* __restrict__ z,
                                                     float* __restrict__ zf) {
    int i = blockIdx.x * 256 + threadIdx.x;          // coalesced read of z
    int w = i & 31;
    int h = (i >> 5) & 31;
    int c = (i >> 10) & 255;
    int b = i >> 18;
    int n = b * 1024 + h * 32 + w;
    zf[n * KDIM + c] = z[i];
}

// ---------------------------------------------------------------------------
// Kernel 2: codebook squared norms, one wave per row
// ---------------------------------------------------------------------------
__global__ __launch_bounds__(256) void emb_norm_kernel(const float* __restrict__ emb,
                                                       float* __restrict__ enorm) {
    int wave = threadIdx.x >> 5, lane = threadIdx.x & 31;
    int row  = blockIdx.x * 8 + wave;
    const float4* p = (const float4*)(emb + (size_t)row * KDIM);
    float s = 0.f;
#pragma unroll
    for (int k = 0; k < 2; ++k) {
        float4 v = p[lane + k * 32];
        s += v.x * v.x + v.y * v.y + v.z * v.z + v.w * v.w;
    }
#pragma unroll
    for (int off = 16; off > 0; off >>= 1) s += __shfl_xor(s, off, 32);
    if (lane == 0) enorm[row] = s;
}

// ---------------------------------------------------------------------------
// Kernel 3: fused GEMM (V_WMMA_F32_16X16X4_F32) + partial argmin.
// Workgroup tile 64(M) x 128(N); 8 waves = 2(M) x 4(N) groups; each wave
// register-blocks 2x2 16x16 tiles so every LDS fragment feeds two WMMAs.
// gridDim.y splits the 16384-code sweep into disjoint halves.
// ---------------------------------------------------------------------------
__global__ __launch_bounds__(256) void vq_argmin_kernel(const float* __restrict__ zf,
                                                        const float* __restrict__ emb,
                                                        const float* __restrict__ enorm,
                                                        float* __restrict__ pVal,
                                                        int* __restrict__ pIdx) {
    __shared__ __align__(16) char smem[AS_BYTES + BS_BYTES];
    float (*As)[ASTRIDE] = (float (*)[ASTRIDE])smem;
    float (*Bs)[ASTRIDE] = (float (*)[ASTRIDE])(smem + AS_BYTES);
    // reduction arrays overlay Bs (used only after a barrier once compute ends)
    float (*redV)[2][8][32] = (float (*)[2][8][32])(smem + AS_BYTES);
    int   (*redI)[2][8][32] = (int   (*)[2][8][32])(smem + AS_BYTES + RED_BYTES);

    const int tid  = threadIdx.x;
    const int wave = tid >> 5;
    const int lane = tid & 31;
    const int l16  = lane & 15;
    const int hi   = lane >> 4;      // K sub-pair / M half selector
    const int mG   = wave & 1;       // 0..1 : M group of 32 rows
    const int nG   = wave >> 1;      // 0..3 : N group of 32 cols
    const int mBase  = blockIdx.x * MT;
    const int chBase = blockIdx.y * CHUNKS_PER_SPLIT;

    // ---- stage A tile (64x256) into LDS, coalesced float4 ----
    {
        const float4* src = (const float4*)(zf + (size_t)mBase * KDIM);
#pragma unroll
        for (int k = 0; k < 16; ++k) {
            int e = tid + k * 256;               // float4 index, 0..4095
            float4 v = src[e];
            *(float4*)&As[e >> 6][(e & 63) * 4] = v;
        }
    }

    float bestV[2][8];
    int   bestI[2][8];
#pragma unroll
    for (int mi = 0; mi < 2; ++mi)
#pragma unroll
        for (int r = 0; r < 8; ++r) { bestV[mi][r] = 3.4e38f; bestI[mi][r] = 0; }

    const float* aBase = &As[mG * 32 + l16][hi * 2];
    const float* bBase = &Bs[nG * 32 + l16][hi * 2];

    v8f Z;
#pragma unroll
    for (int i = 0; i < 8; ++i) Z[i] = 0.f;

    for (int ch = 0; ch < CHUNKS_PER_SPLIT; ++ch) {
        const float4* bsrc =
            (const float4*)(emb + (size_t)(chBase + ch) * NCH * KDIM);

        // first half of chunk: loads issued BEFORE the barrier -> overlap
        // the previous chunk's 2048 WMMAs
        float4 stage[16];
#pragma unroll
        for (int k = 0; k < 16; ++k) stage[k] = bsrc[tid + k * 256];

        __syncthreads();                         // prior chunk's Bs reads done
#pragma unroll
        for (int k = 0; k < 16; ++k) {
            int e = tid + k * 256;               // 0..4095
            *(float4*)&Bs[e >> 6][(e & 63) * 4] = stage[k];
        }
        // second half of chunk
#pragma unroll
        for (int k = 0; k < 16; ++k) stage[k] = bsrc[4096 + tid + k * 256];
#pragma unroll
        for (int k = 0; k < 16; ++k) {
            int e = 4096 + tid + k * 256;        // 4096..8191
            *(float4*)&Bs[e >> 6][(e & 63) * 4] = stage[k];
        }
        __syncthreads();                         // Bs (and on ch==0, As) visible

        v8f C00 = Z, C01 = Z, C10 = Z, C11 = Z;
#pragma unroll 4
        for (int ks = 0; ks < KDIM / 4; ++ks) {  // 64 steps, 4 WMMA each
            v2f a0 = *(const v2f*)(aBase + ks * 4);
            v2f a1 = *(const v2f*)(aBase + 16 * ASTRIDE + ks * 4);
            v2f b0 = *(const v2f*)(bBase + ks * 4);
            v2f b1 = *(const v2f*)(bBase + 16 * ASTRIDE + ks * 4);
            C00 = __builtin_amdgcn_wmma_f32_16x16x4_f32(false, a0, false, b0,
                                                        (short)0, C00, false, false);
            C01 = __builtin_amdgcn_wmma_f32_16x16x4_f32(false, a0, false, b1,
                                                        (short)0, C01, false, false);
            C10 = __builtin_amdgcn_wmma_f32_16x16x4_f32(false, a1, false, b0,
                                                        (short)0, C10, false, false);
            C11 = __builtin_amdgcn_wmma_f32_16x16x4_f32(false, a1, false, b1,
                                                        (short)0, C11, false, false);
        }

        // distance = ||e||^2 - 2 z.e   (||z||^2 constant per row -> dropped)
        int   n0  = (chBase + ch) * NCH + nG * 32 + l16;
        float en0 = enorm[n0];
        float en1 = enorm[n0 + 16];
#pragma unroll
        for (int r = 0; r < 8; ++r) {
            float d;
            d = en0 - 2.f * C00[r]; if (d < bestV[0][r]) { bestV[0][r] = d; bestI[0][r] = n0;      }
            d = en1 - 2.f * C01[r]; if (d < bestV[0][r]) { bestV[0][r] = d; bestI[0][r] = n0 + 16; }
            d = en0 - 2.f * C10[r]; if (d < bestV[1][r]) { bestV[1][r] = d; bestI[1][r] = n0;      }
            d = en1 - 2.f * C11[r]; if (d < bestV[1][r]) { bestV[1][r] = d; bestI[1][r] = n0 + 16; }
        }
    }

    __syncthreads();     // all waves done reading Bs before overlaying it
#pragma unroll
    for (int mi = 0; mi < 2; ++mi)
#pragma unroll
        for (int r = 0; r < 8; ++r) {
            redV[wave][mi][r][lane] = bestV[mi][r];
            redI[wave][mi][r][lane] = bestI[mi][r];
        }
    __syncthreads();

    if (tid < MT) {        // 64 threads: one output row each
        int row = tid;
        int mg = (row >> 5) & 1;     // M group
        int mi = (row >> 4) & 1;     // tile within group
        int g  = (row >> 3) & 1;     // lane group (M half of C layout)
        int r  = row & 7;            // accumulator register
        float bv = 3.4e38f;
        int   bi = 0x7fffffff;
        for (int k = 0; k < 4; ++k) {          // waves sharing this M group
            int w = mg + 2 * k;
            for (int l = 0; l < 16; ++l) {
                float v = redV[w][mi][r][g * 16 + l];
                int   i = redI[w][mi][r][g * 16 + l];
                if (v < bv || (v == bv && i < bi)) { bv = v; bi = i; }
            }
        }
        int o = blockIdx.y * M_TOT + mBase + row;
        pVal[o] = bv;
        pIdx[o] = bi;
    }
}

// ---------------------------------------------------------------------------
// Kernel 3b: merge the NSPLIT partial argmins (split 0 has smaller indices,
// so ties resolve to split 0 == first-min, matching jnp.argmin)
// ---------------------------------------------------------------------------
__global__ __launch_bounds__(256) void argmin_merge_kernel(const float* __restrict__ pVal,
                                                           const int* __restrict__ pIdx,
                                                           int* __restrict__ idx) {
    int m = blockIdx.x * 256 + threadIdx.x;
    float v0 = pVal[m];         int i0 = pIdx[m];
    float v1 = pVal[M_TOT + m]; int i1 = pIdx[M_TOT + m];
    idx[m] = (v1 < v0) ? i1 : i0;
}

// ---------------------------------------------------------------------------
// Kernel 4: gather z_q, write output in (b,c,h,w), per-row loss partials,
// indices as float. One wave per row.
// ---------------------------------------------------------------------------
__global__ __launch_bounds__(256) void gather_loss_kernel(const float* __restrict__ zf,
                                                          const float* __restrict__ emb,
                                                          const int* __restrict__ idx,
                                                          float* __restrict__ zq_out,
                                                          float* __restrict__ idx_out,
                                                          float* __restrict__ partial) {
    int wave = threadIdx.x >> 5, lane = threadIdx.x & 31;
    int n = blockIdx.x * 8 + wave;
    int e = idx[n];
    int b = n >> 10, p = n & 1023;
    const float* zr = zf + (size_t)n * KDIM;
    const float* er = emb + (size_t)e * KDIM;
    float s = 0.f;
#pragma unroll
    for (int k = 0; k < 8; ++k) {
        int c  = lane + k * 32;
        float q = er[c];
        float d = q - zr[c];
        s += d * d;
        zq_out[b * 262144 + c * 1024 + p] = q;   // straight-through value == z_q
    }
#pragma unroll
    for (int off = 16; off > 0; off >>= 1) s += __shfl_xor(s, off, 32);
    if (lane == 0) { partial[n] = s; idx_out[n] = (float)e; }
}

// ---------------------------------------------------------------------------
// Kernel 5: deterministic loss reduction.  loss = (1 + beta) * MSE
// ---------------------------------------------------------------------------
__global__ __launch_bounds__(256) void loss_reduce_kernel(const float* __restrict__ partial,
                                                          float* __restrict__ loss_out) {
    __shared__ float sh[256];
    float s = 0.f;
    for (int k = 0; k < 32; ++k) s += partial[threadIdx.x + k * 256];
    sh[threadIdx.x] = s;
    __syncthreads();
    for (int off = 128; off > 0; off >>= 1) {
        if (threadIdx.x < off) sh[threadIdx.x] += sh[threadIdx.x + off];
        __syncthreads();
    }
    if (threadIdx.x == 0) loss_out[0] = sh[0] * (1.25f / 2097152.0f);
}

// ---------------------------------------------------------------------------
extern "C" void kernel_launch(void* const* d_in, const int* in_sizes, int n_in,
                              void* d_out, int out_size, void* d_ws, size_t ws_size,
                              hipStream_t stream) {
    const float* z   = (const float*)d_in[0];   // [8,256,32,32]
    const float* emb = (const float*)d_in[1];   // [16384,256]
    float* out = (float*)d_out;                 // 524288 zq | 1 loss | 8192 idx

    float* zf      = (float*)d_ws;                      // 8192*256 f32
    float* enorm   = zf + (size_t)M_TOT * KDIM;         // 16384 f32
    int*   idx     = (int*)(enorm + N_TOT);             // 8192 i32
    float* partial = (float*)(idx + M_TOT);             // 8192 f32
    float* pVal    = partial + M_TOT;                   // NSPLIT*8192 f32
    int*   pIdx    = (int*)(pVal + NSPLIT * M_TOT);     // NSPLIT*8192 i32

    pack_z_kernel      <<<M_TOT * KDIM / 256, 256, 0, stream>>>(z, zf);
    emb_norm_kernel    <<<N_TOT / 8,          256, 0, stream>>>(emb, enorm);
    vq_argmin_kernel   <<<dim3(M_TOT / MT, NSPLIT), 256, 0, stream>>>(zf, emb, enorm,
                                                                      pVal, pIdx);
    argmin_merge_kernel<<<M_TOT / 256,        256, 0, stream>>>(pVal, pIdx, idx);
    gather_loss_kernel <<<M_TOT / 8,          256, 0, stream>>>(zf, emb, idx,
                                                                out, out + 524289, partial);
    loss_reduce_kernel <<<1,                  256, 0, stream>>>(partial, out + 524288);
}